// V_l_51204600103593
// MI455X (gfx1250) — compile-verified
//
#include <hip/hip_runtime.h>
#include <math.h>
#include <stdint.h>

// ---------------------------------------------------------------------------
// Kernel 1: zero-fill 512 MiB via CDNA5 async LDS->global streaming stores,
// non-temporal (output is 2.7x L2 and written exactly once). Each thread owns
// a 16 B zero chunk in LDS and streams 64 B per iteration using 4 async
// stores with immediate offsets off one address computation.
// ---------------------------------------------------------------------------
__global__ void fill_zero_async(float* __restrict__ out, size_t nfloats) {
    __shared__ __align__(16) float4 lbuf[256];
    const int tid = threadIdx.x;
    lbuf[tid] = make_float4(0.f, 0.f, 0.f, 0.f);
    __syncthreads();

    // Generic pointer to LDS: low 32 bits are the LDS byte address.
    const uint32_t lds_addr = (uint32_t)(uintptr_t)(&lbuf[tid]);

    float4* __restrict__ out4 = (float4*)out;
    const size_t nchunks = nfloats / 4;   // 16 B chunks
    const size_t ngroups = nchunks / 4;   // 64 B groups (4 chunks)
    const size_t stride  = (size_t)gridDim.x * blockDim.x;
    const size_t t0      = (size_t)blockIdx.x * blockDim.x + tid;

    for (size_t g = t0; g < ngroups; g += stride) {
        const uint64_t gaddr = (uint64_t)(uintptr_t)(out4 + g * 4);
        asm volatile(
            "global_store_async_from_lds_b128 %0, %1, off th:TH_STORE_NT\n"
            "global_store_async_from_lds_b128 %0, %1, off offset:16 th:TH_STORE_NT\n"
            "global_store_async_from_lds_b128 %0, %1, off offset:32 th:TH_STORE_NT\n"
            "global_store_async_from_lds_b128 %0, %1, off offset:48 th:TH_STORE_NT"
            :
            : "v"(gaddr), "v"(lds_addr)
            : "memory");
    }

    // Tail chunks (nchunks % 4) — plain vector stores from the first threads.
    const size_t tail_base = ngroups * 4;
    if (t0 < nchunks - tail_base) {
        out4[tail_base + t0] = make_float4(0.f, 0.f, 0.f, 0.f);
    }
    // Scalar float tail (nfloats % 4); never hit for N=8192.
    if (t0 == 0) {
        for (size_t k = nchunks * 4; k < nfloats; ++k) out[k] = 0.f;
    }

    asm volatile("s_wait_asynccnt 0x0" ::: "memory");
}

// ---------------------------------------------------------------------------
// Kernel 2: diagonal ones + the 4 block-exp entries.
// ---------------------------------------------------------------------------
__global__ void fixup_kernel(const float* __restrict__ bii,
                             const float* __restrict__ bjj,
                             const float* __restrict__ bijr,
                             const float* __restrict__ biji,
                             const int* __restrict__ pci,
                             const int* __restrict__ pcj,
                             const int* __restrict__ pN,
                             float* __restrict__ out) {
    const int N  = *pN;
    const int ci = *pci;
    const int cj = *pcj;
    const int i  = blockIdx.x * blockDim.x + threadIdx.x;

    if (i < N && i != ci && i != cj) {
        // real part of diagonal element (imag already zeroed by fill)
        out[((size_t)i * N + i) * 2] = 1.0f;
    }

    if (i == 0) {
        const float a   = bii[0];
        const float b   = bjj[0];
        const float cr  = bijr[0];
        const float cim = biji[0];

        const float t = 0.5f * (a + b);
        const float d = 0.5f * (a - b);
        const float r = sqrtf(d * d + cr * cr + cim * cim);
        const float sinc = (r > 0.f) ? (sinf(r) / r) : 1.0f;
        const float cosr = cosf(r);
        const float pr = cosf(t);   // phase = e^{i t}
        const float pi = sinf(t);

        // e_ii = phase * (cosr + i*sinc*d)
        const float eii_r = pr * cosr - pi * (sinc * d);
        const float eii_i = pr * (sinc * d) + pi * cosr;
        // e_jj = phase * (cosr - i*sinc*d)
        const float ejj_r = pr * cosr + pi * (sinc * d);
        const float ejj_i = -pr * (sinc * d) + pi * cosr;
        // e_ij = phase * (i*sinc) * (cr + i*cim) = phase * (-sinc*cim + i*sinc*cr)
        const float ur = -sinc * cim, ui = sinc * cr;
        const float eij_r = pr * ur - pi * ui;
        const float eij_i = pr * ui + pi * ur;
        // e_ji = phase * (i*sinc) * (cr - i*cim) = phase * (sinc*cim + i*sinc*cr)
        const float wr = sinc * cim, wi = sinc * cr;
        const float eji_r = pr * wr - pi * wi;
        const float eji_i = pr * wi + pi * wr;

        const size_t ii = ((size_t)ci * N + ci) * 2;
        const size_t jj = ((size_t)cj * N + cj) * 2;
        const size_t ij = ((size_t)ci * N + cj) * 2;
        const size_t ji = ((size_t)cj * N + ci) * 2;
        out[ii] = eii_r; out[ii + 1] = eii_i;
        out[jj] = ejj_r; out[jj + 1] = ejj_i;
        out[ij] = eij_r; out[ij + 1] = eij_i;
        out[ji] = eji_r; out[ji + 1] = eji_i;
    }
}

extern "C" void kernel_launch(void* const* d_in, const int* in_sizes, int n_in,
                              void* d_out, int out_size, void* d_ws, size_t ws_size,
                              hipStream_t stream) {
    (void)in_sizes; (void)n_in; (void)d_ws; (void)ws_size;

    const float* bii  = (const float*)d_in[0];
    const float* bjj  = (const float*)d_in[1];
    const float* bijr = (const float*)d_in[2];
    const float* biji = (const float*)d_in[3];
    const int*   pci  = (const int*)d_in[4];
    const int*   pcj  = (const int*)d_in[5];
    const int*   pN   = (const int*)d_in[6];
    float*       out  = (float*)d_out;

    // out_size = 2*N*N floats (complex64 flattened) -> N host-side, no sync.
    const int N = (int)(sqrt((double)out_size * 0.5) + 0.5);

    // Bulk zero fill: 1024 blocks x 256 threads, 64 B per thread-iteration.
    fill_zero_async<<<1024, 256, 0, stream>>>(out, (size_t)out_size);

    // Diagonal + 2x2 block values (ordered after fill by stream semantics).
    const int blocks = (N + 255) / 256;
    fixup_kernel<<<blocks > 0 ? blocks : 1, 256, 0, stream>>>(
        bii, bjj, bijr, biji, pci, pcj, pN, out);
}